// MyModel_61933428413543
// MI455X (gfx1250) — compile-verified
//
#include <hip/hip_runtime.h>
#include <hip/hip_bf16.h>
#include <cstdint>

// ---------------------------------------------------------------------------
// Net effect of reference: ifftshift(fftshift(x)) == x  (per-axis rolls sum to
// -N ≡ 0), then real/imag re-stacked == x.  => pure identity copy of
// 8*16*320*320*2 f32 = 100 MB.  Bandwidth-bound: ~210 MB traffic / 23.3 TB/s
// ≈ 9 us floor.  Implemented as a B128 streaming copy staged through LDS via
// the CDNA5 async global<->LDS path (ASYNCcnt + s_wait_asynccnt), which keeps
// the payload out of VGPRs and uses the gfx1250-specific data-movement path.
// Phase count F is precomputed on the host to avoid a 64-bit SALU division
// in the kernel prologue.
// ---------------------------------------------------------------------------

#ifndef __has_builtin
#define __has_builtin(x) 0
#endif

#define TPB 256   // 8 wave32 per block
#define KCH 8     // float4 chunks per thread per phase (32 KB LDS staging)

// Exact pointee type expected by the async builtins (per hipcc diagnostic:
// "__attribute__((__vector_size__(4 * sizeof(int)))) int __device__ *").
typedef int v4i __attribute__((__vector_size__(4 * sizeof(int))));
typedef __attribute__((address_space(1))) v4i* gas_v4i;
typedef __attribute__((address_space(3))) v4i* las_v4i;

#if defined(__HIP_DEVICE_COMPILE__) &&                                         \
    __has_builtin(__builtin_amdgcn_global_load_async_to_lds_b128) &&           \
    __has_builtin(__builtin_amdgcn_global_store_async_from_lds_b128)
#define USE_ASYNC_LDS 1
#else
#define USE_ASYNC_LDS 0
#endif

__device__ __forceinline__ void wait_async0() {
#if defined(__HIP_DEVICE_COMPILE__)
#if __has_builtin(__builtin_amdgcn_s_wait_asynccnt)
    __builtin_amdgcn_s_wait_asynccnt(0);
#else
    asm volatile("s_wait_asynccnt 0x0" ::: "memory");
#endif
#endif
}

__global__ void __launch_bounds__(TPB)
identity_copy_kernel(const float* __restrict__ srcf, float* __restrict__ dstf,
                     long long n /* total f32 elements */,
                     int F /* full phases per thread, host-computed */) {
    const long long n4     = n >> 2;                                   // float4 count
    const int       tid    = threadIdx.x;
    const long long gtid   = (long long)blockIdx.x * TPB + tid;
    const long long stride = (long long)gridDim.x * TPB;               // float4 units
    const long long chunk  = stride * KCH;                             // per phase

    const float4* __restrict__ src = (const float4*)srcf;
    float4* __restrict__       dst = (float4*)dstf;

#if USE_ASYNC_LDS
    __shared__ float4 smem[TPB * KCH];                                 // 32 KB
    gas_v4i gsrc = (gas_v4i)srcf;   // C-style cast drops const, retypes to v4i
    gas_v4i gdst = (gas_v4i)dstf;
    las_v4i lds  = (las_v4i)smem;

    for (int f = 0; f < F; ++f) {
        const long long base = (long long)f * chunk + gtid;
        wait_async0();  // previous phase's async stores drained -> LDS reusable
#pragma unroll
        for (int k = 0; k < KCH; ++k) {
            __builtin_amdgcn_global_load_async_to_lds_b128(
                gsrc + (base + (long long)k * stride),
                lds + (k * TPB + tid), 0, 0);
        }
        wait_async0();  // loads have landed in LDS
#pragma unroll
        for (int k = 0; k < KCH; ++k) {
            __builtin_amdgcn_global_store_async_from_lds_b128(
                gdst + (base + (long long)k * stride),
                lds + (k * TPB + tid), 0, 0);
        }
    }
    wait_async0();      // drain final stores
#else
    // Fallback: plain B128 register copy (still saturates HBM).
    for (int f = 0; f < F; ++f) {
        const long long base = (long long)f * chunk + gtid;
#pragma unroll
        for (int k = 0; k < KCH; ++k) {
            const long long i = base + (long long)k * stride;
            dst[i] = src[i];
        }
    }
#endif

    // float4 tail (partial phase)
    for (long long i = (long long)F * chunk + gtid; i < n4; i += stride)
        dst[i] = src[i];

    // scalar tail (n % 4) — zero for this problem, kept for generality
    const long long rem = n & 3LL;
    if (gtid < rem)
        dstf[(n4 << 2) + gtid] = srcf[(n4 << 2) + gtid];
}

extern "C" void kernel_launch(void* const* d_in, const int* in_sizes, int n_in,
                              void* d_out, int out_size, void* d_ws, size_t ws_size,
                              hipStream_t stream) {
    (void)n_in; (void)d_ws; (void)ws_size; (void)in_sizes;

    const float* src = (const float*)d_in[0];
    float*       dst = (float*)d_out;
    const long long n  = (long long)out_size;          // 26,214,400 f32
    const long long n4 = n >> 2;

    const long long per_block = (long long)TPB * KCH;  // float4 per block-phase
    long long blocks64 = (n4 + per_block - 1) / per_block;
    int blocks = (blocks64 > 1600) ? 1600 : (blocks64 < 1 ? 1 : (int)blocks64);

    const long long chunk = (long long)blocks * per_block;  // float4 per grid-phase
    const int F = (int)(n4 / chunk);
    // n4 = 6,553,600; 1600 blocks * 2048 float4/phase -> F = 2, zero tail.

    identity_copy_kernel<<<blocks, TPB, 0, stream>>>(src, dst, n, F);
}